// SupContrastive_21827023798516
// MI455X (gfx1250) — compile-verified
//
#include <hip/hip_runtime.h>
#include <hip/hip_bf16.h>
#include <math.h>

// SupContrastive with one-hot labels reduces exactly to cross-entropy:
//   loss[b] = log( sum_c exp(y_pred[b,c]) ) - y_pred[b, y_true[b]]
//   out     = mean_b loss[b]
//
// Memory-bound: 1.05 GB streamed once; 23.3 TB/s HBM => ~45us floor.
// Kernel 1: one 256-thread block per row, coalesced float4 loads,
//           f32 exp + accumulate, LDS tree reduce, per-row loss -> d_ws.
// Kernel 2: single block reduces 8192 row losses; final 64->16 reduction
//           done with V_WMMA_F32_16X16X4_F32 (B = ones) on wave 0.

typedef __attribute__((ext_vector_type(2))) float v2f;
typedef __attribute__((ext_vector_type(8))) float v8f;

#define B_ROWS 8192
#define C_COLS 32000

__global__ __launch_bounds__(256)
void supcon_row_loss_kernel(const float* __restrict__ y_pred,
                            const int*   __restrict__ y_true,
                            float*       __restrict__ row_loss) {
    __shared__ float sdata[256];

    const int b   = blockIdx.x;
    const int tid = threadIdx.x;

    const float*  row  = y_pred + (size_t)b * C_COLS;
    const float4* row4 = (const float4*)row;          // row stride 128KB -> 16B aligned
    const int     n4   = C_COLS / 4;                  // 8000 float4 per row

    float acc = 0.0f;
    for (int i = tid; i < n4; i += 256) {
        // stay ~16KB ahead of the stream (speculative prefetch; OOB is dropped)
        __builtin_prefetch((const void*)(row4 + i + 1024), 0, 0);
        float4 v = row4[i];
        acc += __expf(v.x) + __expf(v.y) + __expf(v.z) + __expf(v.w);
    }

    sdata[tid] = acc;
    __syncthreads();

    // block tree reduction 256 -> 1
    for (int s = 128; s > 0; s >>= 1) {
        if (tid < s) sdata[tid] += sdata[tid + s];
        __syncthreads();
    }

    if (tid == 0) {
        const int   t  = y_true[b];
        const float pt = row[t];
        row_loss[b] = __logf(sdata[0]) - pt;
    }
}

__global__ __launch_bounds__(256)
void supcon_finalize_kernel(const float* __restrict__ row_loss,
                            float*       __restrict__ out) {
    __shared__ float sdata[256];

    const int tid = threadIdx.x;

    // 8192 -> 256 per-thread partials
    float p = 0.0f;
    for (int i = tid; i < B_ROWS; i += 256) p += row_loss[i];
    sdata[tid] = p;
    __syncthreads();

    // 256 -> 64
    if (tid < 64)
        sdata[tid] = sdata[tid] + sdata[tid + 64] + sdata[tid + 128] + sdata[tid + 192];
    __syncthreads();

    // 64 -> 16 via V_WMMA_F32_16X16X4_F32 with B = ones:
    //   A[m,k] = sdata[m*4 + k]  (16x4),  D[m,n] = sum_k A[m,k]  (replicated over n)
    // Wave-uniform branch: wave 0 fully active (EXEC all ones), waves 1-7 skip.
    if (tid < 32) {
        const int m    = tid & 15;        // A-matrix row for this lane
        const int koff = (tid >> 4) << 1; // lanes 0-15: K=0,1 ; lanes 16-31: K=2,3

        v2f a;
        a.x = sdata[m * 4 + koff + 0];
        a.y = sdata[m * 4 + koff + 1];

        v2f bm;
        bm.x = 1.0f;
        bm.y = 1.0f;

        v8f c = {};
        // 8 args: (neg_a, A, neg_b, B, c_mod, C, reuse_a, reuse_b)
        v8f d = __builtin_amdgcn_wmma_f32_16x16x4_f32(
            false, a, false, bm, (short)0, c, false, false);

        // C/D layout: lane 0 holds D[0..7, 0], lane 16 holds D[8..15, 0]
        float ssum = d[0] + d[1] + d[2] + d[3] + d[4] + d[5] + d[6] + d[7];
        if (tid == 0)  sdata[0] = ssum;   // rows 0..7
        if (tid == 16) sdata[1] = ssum;   // rows 8..15
    }
    __syncthreads();

    if (tid == 0) out[0] = (sdata[0] + sdata[1]) * (1.0f / (float)B_ROWS);
}

extern "C" void kernel_launch(void* const* d_in, const int* in_sizes, int n_in,
                              void* d_out, int out_size, void* d_ws, size_t ws_size,
                              hipStream_t stream) {
    const float* y_pred = (const float*)d_in[0];
    const int*   y_true = (const int*)d_in[1];
    float*       rowbuf = (float*)d_ws;          // 8192 floats of scratch
    float*       outp   = (float*)d_out;

    supcon_row_loss_kernel<<<B_ROWS, 256, 0, stream>>>(y_pred, y_true, rowbuf);
    supcon_finalize_kernel<<<1, 256, 0, stream>>>(rowbuf, outp);
}